// FullAttention_8667244003444
// MI455X (gfx1250) — compile-verified
//
#include <hip/hip_runtime.h>
#include <math.h>

#define N_TOK 3072
#define H_    8
#define D_    24
#define HD_   192     // H*D
#define R_    5
#define K_    8
#define RP1   6       // R+1
#define DH    30      // D + RP1 (augmented dim)
#define DHP   32      // padded to multiple of 4 for WMMA K

// sqrt(log2(e)): pre-scaling q-hat/k-hat by this makes S and the |.|^2 terms
// come out of the WMMA already in log2 units, so E = exp2(S - (q2+k2)/2)
// with no per-element fma -- one v_exp_f32 per kernel entry.
#define SQRT_L2E 1.20112240872f

#if __has_builtin(__builtin_amdgcn_exp2f)
#define EXP2F(x) __builtin_amdgcn_exp2f(x)
#else
#define EXP2F(x) exp2f(x)
#endif

typedef __attribute__((ext_vector_type(2))) float v2f;
typedef __attribute__((ext_vector_type(8))) float v8f;

// ---------------------------------------------------------------------------
// Kernel 1: RPE prep.  qw[h,r] = sum_k exp(clip(sum_d w[h,d,r,k],50));
// s2w[h,j] = sqrt(2*qw_ext[h,j]),  qw_ext = [qw[:,0], qw]   (48 floats)
// ---------------------------------------------------------------------------
__global__ void rpe_prep_kernel(const float* __restrict__ w_rpe,
                                float* __restrict__ s2w) {
  __shared__ float ew[H_ * R_ * K_];
  __shared__ float qw[H_ * R_];
  int t = threadIdx.x;
  if (t < H_ * R_ * K_) {
    int h = t / (R_ * K_);
    int rk = t % (R_ * K_);
    float s = 0.f;
#pragma unroll
    for (int d = 0; d < D_; ++d) s += w_rpe[(h * D_ + d) * (R_ * K_) + rk];
    s = fminf(s, 50.f);
    ew[t] = expf(s);
  }
  __syncthreads();
  if (t < H_ * R_) {
    int h = t / R_, r = t % R_;
    float s = 0.f;
#pragma unroll
    for (int k = 0; k < K_; ++k) s += ew[h * (R_ * K_) + r * K_ + k];
    qw[t] = s;
  }
  __syncthreads();
  if (t < H_ * RP1) {
    int h = t / RP1, j = t % RP1;
    float q = qw[h * R_ + (j == 0 ? 0 : j - 1)];
    s2w[t] = sqrtf(2.f * q);
  }
}

// ---------------------------------------------------------------------------
// Kernel 2: out[n, j] = out_b[j]  (projection bias; heads accumulate on top)
// ---------------------------------------------------------------------------
__global__ void out_init_kernel(const float* __restrict__ out_b,
                                float* __restrict__ out) {
  int i = blockIdx.x * blockDim.x + threadIdx.x;
  if (i < N_TOK * D_) out[i] = out_b[i % D_];
}

// ---------------------------------------------------------------------------
// Kernel 3: fused RBF attention.  One wave32 per (head, 16-row strip).
// QK^T and E@V both via V_WMMA_F32_16X16X4_F32.  Masks folded into the
// exp2 bias as -inf; row sums produced by a ones-column inside the PV WMMA.
// ---------------------------------------------------------------------------
__global__ __launch_bounds__(32)
void rbf_attn_kernel(const float* __restrict__ query,
                     const float* __restrict__ key,
                     const float* __restrict__ value,
                     const int*   __restrict__ pmask,
                     const float* __restrict__ coords,
                     const float* __restrict__ s2w,
                     const float* __restrict__ out_w,
                     float* __restrict__ out) {
  __shared__ float qs[16][DHP];
  __shared__ float ks[16][DHP];
  __shared__ float vs[16][DHP];      // col 24 = 1.0 (rowsum), 25..31 = 0
  __shared__ float es[16][16];
  __shared__ float q2e[16], k2e[16], rs[16];
  __shared__ float attloc[16][D_];

  const int lane = threadIdx.x;
  const int m0 = blockIdx.x * 16;
  const int h  = blockIdx.y;
  const int ml = lane & 15;
  const int kh = (lane >> 4) << 1;   // K sub-offset: 0 (lanes 0-15) / 2 (16-31)

  // ---- load Q-hat tile, pre-scaled by sqrt(log2 e) ----
  for (int idx = lane; idx < 16 * DHP; idx += 32) {
    int m = idx >> 5, c = idx & 31;
    int n = m0 + m;
    float v = 0.f;
    if (c < D_)      v = query[n * HD_ + h * D_ + c];
    else if (c < DH) v = s2w[h * RP1 + (c - D_)] * coords[n * RP1 + (c - D_)];
    qs[m][c] = v * SQRT_L2E;
  }
  __syncthreads();
  if (lane < 16) {
    float s = 0.f;
#pragma unroll
    for (int c = 0; c < DHP; ++c) s += qs[lane][c] * qs[lane][c];
    // -0.5*|q|^2 (log2 units), with the row padding mask folded in as -inf
    q2e[lane] = pmask[m0 + lane] ? (-0.5f * s) : -INFINITY;
  }
  v8f acc0 = {};  // O dims 0..15
  v8f acc1 = {};  // O dims 16..23, col 8 = row sums
  __syncthreads();

  for (int c0 = 0; c0 < N_TOK; c0 += 16) {
    if (c0 + 16 < N_TOK)   // gfx1250 global_prefetch_b8 on next K tile
      __builtin_prefetch(&key[(c0 + 16) * HD_ + h * D_], 0, 1);

    // ---- load K-hat (scaled) and V tiles; V col 24 = 1 for row sums ----
    for (int idx = lane; idx < 16 * DHP; idx += 32) {
      int m = idx >> 5, c = idx & 31;
      int n = c0 + m;
      float kv = 0.f;
      float vv = (c == D_) ? 1.0f : 0.f;
      if (c < D_)      { kv = key[n * HD_ + h * D_ + c];
                         vv = value[n * HD_ + h * D_ + c]; }
      else if (c < DH)   kv = s2w[h * RP1 + (c - D_)] * coords[n * RP1 + (c - D_)];
      ks[m][c] = kv * SQRT_L2E;
      vs[m][c] = vv;
    }
    __syncthreads();
    if (lane < 16) {
      float s = 0.f;
#pragma unroll
      for (int c = 0; c < DHP; ++c) s += ks[lane][c] * ks[lane][c];
      k2e[lane] = pmask[c0 + lane] ? (-0.5f * s) : -INFINITY;
    }
    __syncthreads();

    // ---- S = Q . K^T : 8x V_WMMA_F32_16X16X4_F32 over K=32 ----
    v8f sacc = {};
#pragma unroll
    for (int kc = 0; kc < DHP; kc += 4) {
      v2f a, b;
      a.x = qs[ml][kc + kh];  a.y = qs[ml][kc + kh + 1];
      b.x = ks[ml][kc + kh];  b.y = ks[ml][kc + kh + 1];
      sacc = __builtin_amdgcn_wmma_f32_16x16x4_f32(
          false, a, false, b, (short)0, sacc, false, false);
    }

    // ---- E = exp2(S + bias_row + bias_col); one v_exp_f32 per element ----
    {
      const int nc = lane & 15;
      const int mb = (lane >> 4) * 8;
      const float kb = k2e[nc];
#pragma unroll
      for (int i = 0; i < 8; ++i) {
        int m = mb + i;
        es[m][nc] = EXP2F(sacc[i] + (q2e[m] + kb));
      }
    }
    __syncthreads();

    // ---- [O | rowsum] += E @ [V | 1] : 2 N-tiles x 4 K-chunks of WMMA ----
#pragma unroll
    for (int kc = 0; kc < 16; kc += 4) {
      v2f a, b0, b1;
      a.x  = es[ml][kc + kh];        a.y  = es[ml][kc + kh + 1];
      b0.x = vs[kc + kh][ml];        b0.y = vs[kc + kh + 1][ml];
      b1.x = vs[kc + kh][16 + ml];   b1.y = vs[kc + kh + 1][16 + ml];
      acc0 = __builtin_amdgcn_wmma_f32_16x16x4_f32(
          false, a, false, b0, (short)0, acc0, false, false);
      acc1 = __builtin_amdgcn_wmma_f32_16x16x4_f32(
          false, a, false, b1, (short)0, acc1, false, false);
    }
    __syncthreads();
  }

  // ---- row sums live in acc1 column 8 (v-col 24): broadcast via LDS ----
  if ((lane & 15) == 8) {
    const int mb = (lane >> 4) * 8;
#pragma unroll
    for (int i = 0; i < 8; ++i) rs[mb + i] = acc1[i];
  }
  __syncthreads();

  // ---- normalize, stage head-local attention output ----
  {
    const int nc = lane & 15;
    const int mb = (lane >> 4) * 8;
#pragma unroll
    for (int i = 0; i < 8; ++i) {
      int m = mb + i;
      float inv = 1.f / (rs[m] + 1e-8f);
      attloc[m][nc] = acc0[i] * inv;
      if (nc + 16 < D_) attloc[m][nc + 16] = acc1[i] * inv;
    }
  }
  __syncthreads();

  // ---- fused projection: out[n,j] += sum_d attloc[m][d] * out_w[j, h*D+d] ----
  for (int idx = lane; idx < 16 * D_; idx += 32) {
    int m = idx / D_, j = idx % D_;
    float s = 0.f;
#pragma unroll
    for (int d = 0; d < D_; ++d) s += attloc[m][d] * out_w[j * HD_ + h * D_ + d];
    atomicAdd(&out[(m0 + m) * D_ + j], s);
  }
}

// ---------------------------------------------------------------------------
extern "C" void kernel_launch(void* const* d_in, const int* in_sizes, int n_in,
                              void* d_out, int out_size, void* d_ws, size_t ws_size,
                              hipStream_t stream) {
  const float* query  = (const float*)d_in[0];
  const float* key    = (const float*)d_in[1];
  const float* value  = (const float*)d_in[2];
  const int*   pmask  = (const int*)  d_in[3];
  const float* coords = (const float*)d_in[4];
  const float* w_rpe  = (const float*)d_in[5];
  const float* out_w  = (const float*)d_in[6];
  const float* out_b  = (const float*)d_in[7];
  float* out = (float*)d_out;
  float* s2w = (float*)d_ws;   // 48 floats of scratch

  rpe_prep_kernel<<<1, 384, 0, stream>>>(w_rpe, s2w);
  out_init_kernel<<<(N_TOK * D_ + 255) / 256, 256, 0, stream>>>(out_b, out);
  rbf_attn_kernel<<<dim3(N_TOK / 16, H_), 32, 0, stream>>>(
      query, key, value, pmask, coords, s2w, out_w, out);
}